// EIGNN_scale_w_iter_Broyden_52733608461006
// MI455X (gfx1250) — compile-verified
//
#include <hip/hip_runtime.h>
#include <cmath>
#include <stdint.h>

typedef __attribute__((ext_vector_type(2))) float v2f;
typedef __attribute__((ext_vector_type(8))) float v8f;

#define N_NODES 4096
#define M_FEAT  256
#define GAMMA_C 0.8f
#define EPS_F_C 1e-12f

// ---- CDNA5 async global->LDS copy helpers (portable inline asm; ASYNCcnt) ----
__device__ __forceinline__ uint32_t lds_off(const void* p) {
    // low 32 bits of a generic pointer into LDS == byte offset in LDS window
    return (uint32_t)(uintptr_t)p;
}
__device__ __forceinline__ void async_copy_b128(void* lds_dst, const float* g_src) {
    asm volatile("global_load_async_to_lds_b128 %0, %1, off"
                 :: "v"(lds_off(lds_dst)), "v"(g_src)
                 : "memory");
}
__device__ __forceinline__ void wait_async_le(int) {}
__device__ __forceinline__ void wait_async3() {
    asm volatile("s_wait_asynccnt 0x3" ::: "memory");
}
__device__ __forceinline__ void wait_async0() {
    asm volatile("s_wait_asynccnt 0x0" ::: "memory");
}

// ---------------------------------------------------------------------------
// Tiled fp32 WMMA GEMM:  C[M,N] = op(A) * B   (+ optional gamma*acc + X epilogue)
//   ATRANS=false: A is [M,K] row-major          (C = A*B)
//   ATRANS=true : A is [K,M] row-major          (C = A^T*B)
// Block tile 128(M) x 64(N), K staged 16-deep, DOUBLE-BUFFERED LDS filled by
// global_load_async_to_lds_b128 (3 per thread per chunk, fenced with
// s_wait_asynccnt).  8 wave32 waves; each wave computes a 32x32 tile =
// 2x2 v_wmma_f32_16x16x4_f32 fragments per K-step of 4.
// ---------------------------------------------------------------------------
template<int Mdim, int Ndim, int Kdim, bool ATRANS, bool EPI>
__global__ __launch_bounds__(256)
void wmma_gemm_kernel(const float* __restrict__ A,
                      const float* __restrict__ B,
                      float* __restrict__ C,
                      const float* __restrict__ Xadd)
{
    // A tile: NN -> M-major [128][20] (A-frag pair = one aligned b64 read)
    //         TN -> K-major [16][132] (global side contiguous for async copy)
    constexpr int AH = ATRANS ? 16  : 128;
    constexpr int AW = ATRANS ? 132 : 20;
    __shared__ float As[2][AH][AW];
    __shared__ float Bs[2][16][68];   // K-major B tile (row stride 272B)

    const int m0   = blockIdx.y * 128;
    const int n0   = blockIdx.x * 64;
    const int tid  = threadIdx.x;
    const int lane = tid & 31;
    const int wave = tid >> 5;
    const int wm   = wave >> 1;        // 0..3  (M direction)
    const int wn   = wave & 1;         // 0..1  (N direction)
    const int l16  = lane & 15;
    const int kb   = (lane >> 4) << 1; // lanes 16-31 hold K+2,K+3
    const int mb   = (lane >> 4) << 3; // C/D rows 8..15 in lanes 16-31

    // per-thread async staging coordinates (3 x b128 per chunk per thread)
    const int a0_row = tid >> 2;            // NN: rows 0..63   (i=0)
    const int a1_row = (tid + 256) >> 2;    // NN: rows 64..127 (i=1)
    const int a_kq   = (tid & 3) << 2;      // NN: k quad 0/4/8/12
    const int at_kr0 = tid >> 5;            // TN: k rows 0..7
    const int at_kr1 = (tid + 256) >> 5;    // TN: k rows 8..15
    const int at_mq  = (tid & 31) << 2;     // TN: m quad
    const int b_kr   = tid >> 4;            // B: k row 0..15
    const int b_nq   = (tid & 15) << 2;     // B: n quad

    auto issue_chunk = [&](int buf, int k0) {
        if constexpr (!ATRANS) {
            async_copy_b128(&As[buf][a0_row][a_kq],
                            A + (size_t)(m0 + a0_row) * Kdim + k0 + a_kq);
            async_copy_b128(&As[buf][a1_row][a_kq],
                            A + (size_t)(m0 + a1_row) * Kdim + k0 + a_kq);
        } else {
            async_copy_b128(&As[buf][at_kr0][at_mq],
                            A + (size_t)(k0 + at_kr0) * Mdim + m0 + at_mq);
            async_copy_b128(&As[buf][at_kr1][at_mq],
                            A + (size_t)(k0 + at_kr1) * Mdim + m0 + at_mq);
        }
        async_copy_b128(&Bs[buf][b_kr][b_nq],
                        B + (size_t)(k0 + b_kr) * Ndim + n0 + b_nq);
    };

    v8f acc[2][2] = {};

    issue_chunk(0, 0);
    for (int k0 = 0; k0 < Kdim; k0 += 16) {
        const int  buf  = (k0 >> 4) & 1;
        const bool more = (k0 + 16) < Kdim;
        if (more) issue_chunk(buf ^ 1, k0 + 16);   // overlap next chunk
        if (more) wait_async3();                   // current chunk resident
        else      wait_async0();
        __syncthreads();

        #pragma unroll
        for (int ks = 0; ks < 16; ks += 4) {
            v2f af[2], bf[2];
            #pragma unroll
            for (int mi = 0; mi < 2; ++mi) {
                const int m = wm * 32 + mi * 16 + l16;
                if constexpr (!ATRANS) {
                    af[mi] = *(const v2f*)&As[buf][m][ks + kb];   // ds_load_b64
                } else {
                    af[mi].x = As[buf][ks + kb + 0][m];
                    af[mi].y = As[buf][ks + kb + 1][m];
                }
            }
            #pragma unroll
            for (int ni = 0; ni < 2; ++ni) {
                const int n = wn * 32 + ni * 16 + l16;
                bf[ni].x = Bs[buf][ks + kb + 0][n];
                bf[ni].y = Bs[buf][ks + kb + 1][n];
            }
            #pragma unroll
            for (int mi = 0; mi < 2; ++mi)
                #pragma unroll
                for (int ni = 0; ni < 2; ++ni)
                    acc[mi][ni] = __builtin_amdgcn_wmma_f32_16x16x4_f32(
                        false, af[mi], false, bf[ni],
                        (short)0, acc[mi][ni], false, false);
        }
        __syncthreads();   // all reads of buf done before it is refilled
    }

    // ---- epilogue ----
    #pragma unroll
    for (int mi = 0; mi < 2; ++mi) {
        #pragma unroll
        for (int ni = 0; ni < 2; ++ni) {
            const int gn  = n0 + wn * 32 + ni * 16 + l16;
            const int gmb = m0 + wm * 32 + mi * 16 + mb;
            #pragma unroll
            for (int r = 0; r < 8; ++r) {
                const size_t idx = (size_t)(gmb + r) * Ndim + gn;
                const float v = acc[mi][ni][r];
                if constexpr (EPI) C[idx] = GAMMA_C * v + Xadd[idx];
                else               C[idx] = v;
            }
        }
    }
}

// FF = F^T F  (256x256x256, tiny; bitwise-symmetric by identical k-order)
__global__ __launch_bounds__(256)
void ff_kernel(const float* __restrict__ F, float* __restrict__ FF)
{
    const int i = blockIdx.x;
    const int j = threadIdx.x;
    float acc = 0.f;
    for (int k = 0; k < M_FEAT; ++k)
        acc += F[k * M_FEAT + i] * F[k * M_FEAT + j];
    FF[i * M_FEAT + j] = acc;
}

// Deterministic single-block Frobenius-norm reduction of FF (65536 elems)
__global__ __launch_bounds__(256)
void norm_kernel(const float* __restrict__ FF, float* __restrict__ nrm)
{
    __shared__ float s[256];
    float acc = 0.f;
    for (int idx = threadIdx.x; idx < M_FEAT * M_FEAT; idx += 256) {
        const float v = FF[idx];
        acc += v * v;
    }
    s[threadIdx.x] = acc;
    __syncthreads();
    for (int off = 128; off > 0; off >>= 1) {
        if (threadIdx.x < (unsigned)off) s[threadIdx.x] += s[threadIdx.x + off];
        __syncthreads();
    }
    if (threadIdx.x == 0) nrm[0] = sqrtf(s[0]);
}

// G = FF / (norm + eps_F)
__global__ __launch_bounds__(256)
void scale_kernel(const float* __restrict__ FF, const float* __restrict__ nrm,
                  float* __restrict__ G)
{
    const int i = blockIdx.x * 256 + threadIdx.x;
    G[i] = FF[i] / (nrm[0] + EPS_F_C);
}

// Z0 = X  (f(0) = X, so the 51 total applications become 50 starting from X)
__global__ __launch_bounds__(256)
void copy4_kernel(const float4* __restrict__ src, float4* __restrict__ dst)
{
    const int i = blockIdx.x * 256 + threadIdx.x;
    dst[i] = src[i];
}

extern "C" void kernel_launch(void* const* d_in, const int* in_sizes, int n_in,
                              void* d_out, int out_size, void* d_ws, size_t ws_size,
                              hipStream_t stream)
{
    (void)in_sizes; (void)n_in; (void)out_size; (void)ws_size;
    const float* X   = (const float*)d_in[0];   // [4096, 256]
    const float* adj = (const float*)d_in[1];   // [4096, 4096]
    const float* F   = (const float*)d_in[2];   // [256, 256]
    float* out = (float*)d_out;                 // [4096, 256]

    // workspace layout (P:64MB | Za:4MB | T:4MB | FF:256KB | G:256KB | norm)
    float* P   = (float*)d_ws;
    float* Za  = P  + (size_t)N_NODES * N_NODES;
    float* T   = Za + (size_t)N_NODES * M_FEAT;
    float* FFb = T  + (size_t)N_NODES * M_FEAT;
    float* Gb  = FFb + M_FEAT * M_FEAT;
    float* nrm = Gb  + M_FEAT * M_FEAT;

    const dim3 blk(256);

    // P = adj @ adj   (Sk; the transpose in St is absorbed by reading P
    // K-major in the TN GEMM below)
    const dim3 g1(N_NODES / 64, N_NODES / 128);
    wmma_gemm_kernel<N_NODES, N_NODES, N_NODES, false, false>
        <<<g1, blk, 0, stream>>>(adj, adj, P, nullptr);

    // G = g(F); G^T == G bitwise (symmetric construction)
    ff_kernel  <<<dim3(M_FEAT), blk, 0, stream>>>(F, FFb);
    norm_kernel<<<dim3(1),      blk, 0, stream>>>(FFb, nrm);
    scale_kernel<<<dim3(M_FEAT * M_FEAT / 256), blk, 0, stream>>>(FFb, nrm, Gb);

    // Z0 = X in d_out (ping-pong partner is Za)
    copy4_kernel<<<dim3(N_NODES * M_FEAT / 4 / 256), blk, 0, stream>>>(
        (const float4*)X, (float4*)out);

    // 50 Picard applications of f(Z) = gamma*(Sk^T Z)G + X; last lands in d_out
    const dim3 g2(M_FEAT / 64, N_NODES / 128);
    for (int it = 0; it < 50; ++it) {
        const float* Zin = (it & 1) ? Za : out;
        float*      Zout = (it & 1) ? out : Za;
        // T = P^T @ Zin  (TN: stage P K-major, contiguous async row copies)
        wmma_gemm_kernel<N_NODES, M_FEAT, N_NODES, true, false>
            <<<g2, blk, 0, stream>>>(P, Zin, T, nullptr);
        // Zout = gamma*(T @ G) + X
        wmma_gemm_kernel<N_NODES, M_FEAT, M_FEAT, false, true>
            <<<g2, blk, 0, stream>>>(T, Gb, Zout, X);
    }
}